// PLILayer_65962107732080
// MI455X (gfx1250) — compile-verified
//
#include <hip/hip_runtime.h>
#include <hip/hip_bf16.h>

// ---------------------------------------------------------------------------
// PLI via closed-form Hilbert circulant GEMM (V_WMMA_F32_16X16X4_F32) + sign sum
//   x        : [16, 64, 2048] f32  == X [1024, 2048]
//   y=H(x)   : Y = X @ W, W[m,n] = ghat[(n-m) mod 2048]  (exact FFT-Hilbert kernel)
//   pli[b,i,j] = | sum_s sign(y_i x_j - x_i y_j) | / 2048
// ---------------------------------------------------------------------------

#define SLEN 2048
#define NROWS 1024   // 16*64 channels
#define PI_D 3.14159265358979323846

typedef __attribute__((ext_vector_type(2))) float v2f;
typedef __attribute__((ext_vector_type(8))) float v8f;

// ---- Kernel 1: closed-form discrete Hilbert kernel ------------------------
// ghat[n] = (2/S) * cot(pi*n/S) for odd n, else 0   (S even)
__global__ void pli_ghat_kernel(float* __restrict__ ghat) {
    int n = blockIdx.x * blockDim.x + threadIdx.x;
    if (n >= SLEN) return;
    float v = 0.0f;
    if (n & 1) {
        double t = tan(PI_D * (double)n / (double)SLEN);
        v = (float)((2.0 / (double)SLEN) / t);
    }
    ghat[n] = v;
}

// ---- Kernel 2: Y[1024,2048] = X @ W via WMMA f32 16x16x4 ------------------
// One wave computes a 16x64 output strip (4 accumulators). The circulant
// generator is stored REVERSED and duplicated 2x in LDS:
//     gh2r[t] = ghat[(2047 - t) & 2047]
// so each B operand pair loads in WMMA component order {b.x, b.y} from one
// linearly-incrementing base (no modular AND, no component swap movs); the 4
// strip loads use immediate DS offsets 0/16/32/48. 2048 full waves (EXEC
// all-ones at every WMMA, as required).
__global__ void pli_hilbert_wmma_kernel(const float* __restrict__ x,
                                        const float* __restrict__ ghat,
                                        float* __restrict__ y) {
    __shared__ float gh2r[2 * SLEN];
    for (int t = threadIdx.x; t < 2 * SLEN; t += blockDim.x)
        gh2r[t] = ghat[(SLEN - 1 - t) & (SLEN - 1)];
    __syncthreads();

    const int wave = threadIdx.x >> 5;
    const int lane = threadIdx.x & 31;
    const int tile = blockIdx.x * (blockDim.x >> 5) + wave;
    const int ntiles_n = SLEN / 64;              // 32 strips of width 64
    const int m0 = (tile / ntiles_n) * 16;
    const int n0 = (tile % ntiles_n) * 64;

    const int g  = lane >> 4;   // lane group: K pair selector
    const int lr = lane & 15;   // A: row M ; B: col N

    v8f acc0 = {}, acc1 = {}, acc2 = {}, acc3 = {};
    const float* xrow = x + (size_t)(m0 + lr) * SLEN;

    for (int k = 0; k < SLEN; k += 4) {
        const int ka = k + 2 * g;
        // A tile 16x4: lane(lr,g) holds X[m0+lr, ka+{0,1}]
        v2f a;
        a.x = xrow[ka];
        a.y = xrow[ka + 1];
        // B tiles 4x16 at n = n0 + 16q:
        //   p_q = ka - lr - (n0+16q) + 2047;  gh2r[p_q]   = ghat[(n+lr-ka  )&2047] = b.x
        //                                     gh2r[p_q+1] = ghat[(n+lr-ka-1)&2047] = b.y
        // base at q=3, strips at immediate offsets +48/+32/+16/+0
        const float* gb = gh2r + (ka - lr - n0 + (SLEN - 1) - 48);  // >= 0
        v2f b0, b1, b2, b3;
        b3.x = gb[0];  b3.y = gb[1];
        b2.x = gb[16]; b2.y = gb[17];
        b1.x = gb[32]; b1.y = gb[33];
        b0.x = gb[48]; b0.y = gb[49];
        acc0 = __builtin_amdgcn_wmma_f32_16x16x4_f32(false, a, false, b0, (short)0, acc0, false, false);
        acc1 = __builtin_amdgcn_wmma_f32_16x16x4_f32(false, a, false, b1, (short)0, acc1, false, false);
        acc2 = __builtin_amdgcn_wmma_f32_16x16x4_f32(false, a, false, b2, (short)0, acc2, false, false);
        acc3 = __builtin_amdgcn_wmma_f32_16x16x4_f32(false, a, false, b3, (short)0, acc3, false, false);
    }

    // C/D layout: VGPR r -> M = m0 + r + 8*g, N = n0 + lr (+16 per strip)
    float* yb = y + (size_t)(m0 + 8 * g) * SLEN + (n0 + lr);
#pragma unroll
    for (int r = 0; r < 8; ++r) {
        yb[(size_t)r * SLEN + 0]  = acc0[r];
        yb[(size_t)r * SLEN + 16] = acc1[r];
        yb[(size_t)r * SLEN + 32] = acc2[r];
        yb[(size_t)r * SLEN + 48] = acc3[r];
    }
}

// ---- Kernel 3: wave-per-pair sign reduction -------------------------------
__global__ void pli_pairs_kernel(const float* __restrict__ x,
                                 const float* __restrict__ y,
                                 float* __restrict__ out) {
    const int wid  = (int)((blockIdx.x * blockDim.x + threadIdx.x) >> 5);
    const int lane = threadIdx.x & 31;
    // wid in [0, 16*64*64)
    const int b = wid >> 12;
    const int i = (wid >> 6) & 63;
    const int j = wid & 63;

    const float* xi = x + (size_t)(b * 64 + i) * SLEN;
    const float* yi = y + (size_t)(b * 64 + i) * SLEN;
    const float* xj = x + (size_t)(b * 64 + j) * SLEN;
    const float* yj = y + (size_t)(b * 64 + j) * SLEN;

    float sum = 0.0f;
    for (int s = lane; s < SLEN; s += 32) {
        const float v = yi[s] * xj[s] - xi[s] * yj[s];
        sum += (v > 0.0f) ? 1.0f : ((v < 0.0f) ? -1.0f : 0.0f);
    }
    // wave32 butterfly reduction
#pragma unroll
    for (int off = 16; off > 0; off >>= 1)
        sum += __shfl_xor(sum, off, 32);

    if (lane == 0) out[wid] = fabsf(sum) * (1.0f / (float)SLEN);
}

extern "C" void kernel_launch(void* const* d_in, const int* in_sizes, int n_in,
                              void* d_out, int out_size, void* d_ws, size_t ws_size,
                              hipStream_t stream) {
    (void)in_sizes; (void)n_in; (void)out_size; (void)ws_size;
    const float* x   = (const float*)d_in[0];
    float*       out = (float*)d_out;

    float* ghat = (float*)d_ws;              // 2048 floats (8 KB)
    float* y    = (float*)d_ws + SLEN;       // 1024*2048 floats (8 MB)

    // 1) closed-form Hilbert circulant generator
    pli_ghat_kernel<<<SLEN / 256, 256, 0, stream>>>(ghat);

    // 2) Y = X @ W via WMMA f32: 64 m-tiles * 32 n-strips = 2048 waves
    pli_hilbert_wmma_kernel<<<(NROWS / 16) * (SLEN / 64) / 8, 256, 0, stream>>>(x, ghat, y);

    // 3) PLI: one wave per (b,i,j) pair -> 65536 waves
    pli_pairs_kernel<<<16 * 64 * 64 / 8, 256, 0, stream>>>(x, y, out);
}